// lambdaLinear_39264591020649
// MI455X (gfx1250) — compile-verified
//
#include <hip/hip_runtime.h>

#define ORDER   8
#define NCOEF   (ORDER + 1)
#define TBL     368          // covers doy 0..365, padded to multiple of 4 dwords
#define DAYS_F  365.0f

typedef __attribute__((ext_vector_type(4))) int          vi4;
typedef __attribute__((ext_vector_type(4))) float        vf4;
typedef __attribute__((ext_vector_type(4))) unsigned int u32x4;
typedef __attribute__((ext_vector_type(4))) int          i32x4;
typedef __attribute__((ext_vector_type(8))) int          i32x8;

// ---------------------------------------------------------------------------
// Kernel 1: poly(doy) lookup table -> d_ws (368 floats, 1.47 KB).
// Horner evaluation; runs once per launch, negligible cost.
// ---------------------------------------------------------------------------
__global__ void lambda_build_table_kernel(const float* __restrict__ coefs,
                                          float* __restrict__ table) {
    int d = threadIdx.x;
    if (d < TBL) {
        float t   = (float)d * (1.0f / DAYS_F);
        float acc = coefs[ORDER];
#pragma unroll
        for (int k = ORDER - 1; k >= 0; --k) acc = fmaf(acc, t, coefs[k]);
        table[d] = acc;
    }
}

// ---------------------------------------------------------------------------
// Kernel 2: HBM-streaming evaluation.
//   - wave 0 DMAs the 1.47KB table into LDS via the Tensor Data Mover
//   - grid-stride float4/int4 non-temporal streaming of doy/promo -> out
//   - per element: 1 LDS gather + 1 FMA + 1 select
// ---------------------------------------------------------------------------
__global__ __launch_bounds__(256) void lambda_stream_kernel(
    const int*   __restrict__ doy,
    const float* __restrict__ promo,
    const float* __restrict__ gtable,
    const float* __restrict__ promotion_c,
    float*       __restrict__ out,
    int n) {
    __shared__ float tbl[TBL];

#if defined(__gfx1250__) && __has_builtin(__builtin_amdgcn_tensor_load_to_lds)
    if (threadIdx.x < 32) {  // one full wave issues the TDM op (EXEC all ones)
        unsigned long long ga = (unsigned long long)(size_t)gtable;
        unsigned lds_off      = (unsigned)(size_t)&tbl[0];  // low 32b = LDS byte offset

        // D# group 0: count=1 | lds_addr | global_addr[56:0] | type=2
        u32x4 g0;
        g0.x = 1u;                                            // count = 1 (valid)
        g0.y = lds_off;                                       // lds_addr (bytes)
        g0.z = (unsigned)(ga & 0xffffffffull);                // global_addr[31:0]
        g0.w = (unsigned)((ga >> 32) & 0x01ffffffull)         // global_addr[56:32]
             | (2u << 30);                                    // type = 2 ("image")

        // D# group 1: data_size=4B, tensor 368x1, tile 368x1, stride 368
        i32x8 g1;
        g1[0] = (2 << 16);          // data_size = 2 -> 4 bytes; wg_mask = 0
        g1[1] = (TBL << 16);        // tensor_dim0[15:0] in [47:32+16]
        g1[2] = (1 << 16);          // tensor_dim0 hi = 0 | tensor_dim1 = 1
        g1[3] = (TBL << 16);        // tensor_dim1 hi = 0 | tile_dim0 = 368
        g1[4] = 1;                  // tile_dim1 = 1, tile_dim2 = 0
        g1[5] = TBL;                // tensor_dim0_stride[31:0]
        g1[6] = 0;
        g1[7] = 0;

        i32x4 z4 = {0, 0, 0, 0};
#if __clang_major__ >= 23
        i32x8 z8 = {0, 0, 0, 0, 0, 0, 0, 0};
        __builtin_amdgcn_tensor_load_to_lds(g0, g1, z4, z4, z8, 0);
#else
        __builtin_amdgcn_tensor_load_to_lds(g0, g1, z4, z4, 0);
#endif
        __builtin_amdgcn_s_wait_tensorcnt(0);
    }
#else
    for (int i = threadIdx.x; i < TBL; i += blockDim.x) tbl[i] = gtable[i];
#endif
    __syncthreads();

    const float pc = promotion_c[0];          // uniform -> s_load

    const long long n4     = (long long)(n >> 2);
    const long long tid    = (long long)blockIdx.x * blockDim.x + threadIdx.x;
    const long long stride = (long long)gridDim.x * blockDim.x;

    const vi4* __restrict__ d4 = (const vi4*)doy;
    const vf4* __restrict__ p4 = (const vf4*)promo;
    vf4*       __restrict__ o4 = (vf4*)out;

    for (long long i = tid; i < n4; i += stride) {
        vi4 d = __builtin_nontemporal_load(d4 + i);   // global_load_b128 NT
        vf4 p = __builtin_nontemporal_load(p4 + i);   // global_load_b128 NT
        vf4 r;
        r.x = (d.x == 0) ? 0.0f : fmaf(pc, p.x, tbl[d.x]);
        r.y = (d.y == 0) ? 0.0f : fmaf(pc, p.y, tbl[d.y]);
        r.z = (d.z == 0) ? 0.0f : fmaf(pc, p.z, tbl[d.z]);
        r.w = (d.w == 0) ? 0.0f : fmaf(pc, p.w, tbl[d.w]);
        __builtin_nontemporal_store(r, o4 + i);       // global_store_b128 NT
    }

    // scalar tail for n not divisible by 4 (N=16.7M is, but stay general)
    const long long base = n4 << 2;
    const long long rem  = (long long)n - base;
    if (tid < rem) {
        int dd = doy[base + tid];
        out[base + tid] = (dd == 0) ? 0.0f : fmaf(pc, promo[base + tid], tbl[dd]);
    }
}

// ---------------------------------------------------------------------------
// Host-side launcher. Inputs (setup_inputs order):
//   d_in[0] day_of_year : int32  [N]
//   d_in[1] onpromotion : float32[N]
//   d_in[2] coefs       : float32[9]
//   d_in[3] promotion_c : float32[1]
// d_out: float32[N].  d_ws: first 1.47 KB used for the poly table.
// ---------------------------------------------------------------------------
extern "C" void kernel_launch(void* const* d_in, const int* in_sizes, int n_in,
                              void* d_out, int out_size, void* d_ws, size_t ws_size,
                              hipStream_t stream) {
    const int*   doy   = (const int*)d_in[0];
    const float* promo = (const float*)d_in[1];
    const float* coefs = (const float*)d_in[2];
    const float* pc    = (const float*)d_in[3];
    float*       out   = (float*)d_out;
    float*       table = (float*)d_ws;
    const int    n     = in_sizes[0];

    hipLaunchKernelGGL(lambda_build_table_kernel, dim3(1), dim3(TBL), 0, stream,
                       coefs, table);

    // 2048 blocks x 256 threads = 16K wave32s; ~8 float4 iterations per thread.
    hipLaunchKernelGGL(lambda_stream_kernel, dim3(2048), dim3(256), 0, stream,
                       doy, promo, table, pc, out, n);
}